// GNNModel_5007931867539
// MI455X (gfx1250) — compile-verified
//
#include <hip/hip_runtime.h>

typedef __attribute__((ext_vector_type(2))) float v2f;
typedef __attribute__((ext_vector_type(8))) float v8f;

// ---------------------------------------------------------------------------
// WMMA fp32 GEMM:  C[M,N] = A[M,K] * B[K,N], all row-major.
// M multiple of 16, N multiple of 64, K multiple of 4 (true for this model).
// Block = 128 threads (4 waves). grid = (M/16, N/64). Each wave owns one
// 16x16 output tile; the 16xK A tile is shared through LDS (padded stride
// K+1 -> conflict-free: row stride 129/257 mod 64 banks is odd).
// ---------------------------------------------------------------------------
template <int K>
__global__ __launch_bounds__(128) void gemm_wmma_f32(
    const float* __restrict__ A, const float* __restrict__ B,
    float* __restrict__ C, int N) {
  constexpr int KP = K + 1;
  __shared__ float As[16 * KP];

  const int m0 = blockIdx.x * 16;
  // Cooperative A-tile load (uniform trip count -> no divergence).
  for (int i = threadIdx.x; i < 16 * K; i += 128) {
    int r = i / K, c = i - r * K;
    As[r * KP + c] = A[(size_t)(m0 + r) * K + c];
  }
  __syncthreads();

  const int wave = threadIdx.x >> 5;
  const int lane = threadIdx.x & 31;
  const int n0 = blockIdx.y * 64 + wave * 16;

  const int mrow  = lane & 15;        // A: row within tile
  const int khalf = (lane >> 4) * 2;  // A/B: lanes 16-31 hold K+2,K+3
  const int ccol  = n0 + (lane & 15);

  v8f acc = {0.f, 0.f, 0.f, 0.f, 0.f, 0.f, 0.f, 0.f};

#pragma unroll 4
  for (int k = 0; k < K; k += 4) {
    v2f a, b;
    a.x = As[mrow * KP + k + khalf + 0];
    a.y = As[mrow * KP + k + khalf + 1];
    const float* bp = B + (size_t)(k + khalf) * N + ccol;
    b.x = bp[0];
    b.y = bp[N];
    acc = __builtin_amdgcn_wmma_f32_16x16x4_f32(
        false, a, false, b, (short)0, acc, false, false);
  }

  // C/D layout: VGPR r -> (row r, lanes 0-15) / (row r+8, lanes 16-31)
  const int crow = m0 + (lane >> 4) * 8;
#pragma unroll
  for (int r = 0; r < 8; ++r)
    C[(size_t)(crow + r) * N + ccol] = acc[r];
}

// ---------------------------------------------------------------------------
// Per-node attention logits: es[n,h] = sum_c h[n,h*C+c]*a_src[h,c]  (H=2)
// One wave per node, shuffle reduction.
// ---------------------------------------------------------------------------
__global__ void attn_logits(const float* __restrict__ hmat,
                            const float* __restrict__ a_s,
                            const float* __restrict__ a_d,
                            float* __restrict__ es, float* __restrict__ ed,
                            int N, int C) {
  int node = blockIdx.x * (blockDim.x >> 5) + (threadIdx.x >> 5);
  int lane = threadIdx.x & 31;
  if (node >= N) return;
  const int HC = 2 * C;
#pragma unroll
  for (int h = 0; h < 2; ++h) {
    float ss = 0.f, sd = 0.f;
    for (int c = lane; c < C; c += 32) {
      float v = hmat[(size_t)node * HC + h * C + c];
      ss += v * a_s[h * C + c];
      sd += v * a_d[h * C + c];
    }
#pragma unroll
    for (int off = 16; off > 0; off >>= 1) {
      ss += __shfl_down(ss, off, 32);
      sd += __shfl_down(sd, off, 32);
    }
    if (lane == 0) {
      es[node * 2 + h] = ss;
      ed[node * 2 + h] = sd;
    }
  }
}

// ---- order-preserving float<->uint key for atomic max (order-independent) --
__device__ __forceinline__ unsigned fkey(float f) {
  unsigned u = __float_as_uint(f);
  return (u & 0x80000000u) ? ~u : (u | 0x80000000u);
}
__device__ __forceinline__ float funkey(unsigned k) {
  return __uint_as_float((k & 0x80000000u) ? (k & 0x7FFFFFFFu) : ~k);
}

__global__ void init_stats(unsigned* __restrict__ mkey, float* __restrict__ ssum,
                           int n2) {
  int t = blockIdx.x * blockDim.x + threadIdx.x;
  if (t >= n2) return;
  mkey[t] = 0x007FFFFFu;  // fkey(-inf)
  ssum[t] = 0.f;
}

template <int HC>
__global__ void init_bias(float* __restrict__ acc, const float* __restrict__ b,
                          long long total) {
  long long t = (long long)blockIdx.x * blockDim.x + threadIdx.x;
  if (t >= total) return;
  acc[t] = b[(int)(t & (HC - 1))];
}

// Pass 1: e = leaky_relu(es[src]+ed[dst]); segment max over dst (ordered-key).
__global__ void edge_max(const int* __restrict__ ei, int E, int N,
                         const float* __restrict__ es, const float* __restrict__ ed,
                         float* __restrict__ eedge, unsigned* __restrict__ mkey) {
  int t = blockIdx.x * blockDim.x + threadIdx.x;
  if (t >= E + N) return;
  int s, d;
  if (t < E) { s = ei[t]; d = ei[E + t]; } else { s = d = t - E; }
#pragma unroll
  for (int h = 0; h < 2; ++h) {
    float e = es[s * 2 + h] + ed[d * 2 + h];
    e = e > 0.f ? e : 0.2f * e;
    eedge[(size_t)t * 2 + h] = e;
    atomicMax(&mkey[d * 2 + h], fkey(e));
  }
}

// Pass 2: w = exp(e - m[dst]); segment sum over dst.
__global__ void edge_exp(const int* __restrict__ ei, int E, int N,
                         const unsigned* __restrict__ mkey,
                         float* __restrict__ eedge,  // in: e, out: w
                         float* __restrict__ ssum) {
  int t = blockIdx.x * blockDim.x + threadIdx.x;
  if (t >= E + N) return;
  int d = (t < E) ? ei[E + t] : (t - E);
#pragma unroll
  for (int h = 0; h < 2; ++h) {
    float w = expf(eedge[(size_t)t * 2 + h] - funkey(mkey[d * 2 + h]));
    eedge[(size_t)t * 2 + h] = w;
    atomicAdd(&ssum[d * 2 + h], w);
  }
}

// Pass 3 (bandwidth dominant): acc[dst,:] += alpha * h[src,:].
// One thread per (edge, channel): h[src] reads fully coalesced per wave.
template <int HC>
__global__ void edge_scatter(const int* __restrict__ ei, int E, int N,
                             const float* __restrict__ w,
                             const float* __restrict__ ssum,
                             const float* __restrict__ hmat,
                             float* __restrict__ acc) {
  long long t = (long long)blockIdx.x * blockDim.x + threadIdx.x;
  long long total = (long long)(E + N) * HC;
  if (t >= total) return;
  int idx = (int)(t & (HC - 1));
  int e = (int)(t / HC);
  int s, d;
  if (e < E) { s = ei[e]; d = ei[E + e]; } else { s = d = e - E; }
  int h = idx / (HC / 2);
  float alpha = w[(size_t)e * 2 + h] / (ssum[d * 2 + h] + 1e-16f);
  atomicAdd(&acc[(size_t)d * HC + idx],
            alpha * hmat[(size_t)s * HC + idx]);
}

// BN(eval) + ReLU, elementwise in place.
template <int HC>
__global__ void bn_relu(float* __restrict__ io, const float* __restrict__ g,
                        const float* __restrict__ be, const float* __restrict__ rm,
                        const float* __restrict__ rv, long long total) {
  long long t = (long long)blockIdx.x * blockDim.x + threadIdx.x;
  if (t >= total) return;
  int j = (int)(t & (HC - 1));
  float y = (io[t] - rm[j]) * rsqrtf(rv[j] + 1e-5f) * g[j] + be[j];
  io[t] = y > 0.f ? y : 0.f;
}

// ---------------------------------------------------------------------------
extern "C" void kernel_launch(void* const* d_in, const int* in_sizes, int n_in,
                              void* d_out, int out_size, void* d_ws, size_t ws_size,
                              hipStream_t stream) {
  const float* x   = (const float*)d_in[0];
  const int*   ei  = (const int*)d_in[1];
  const float* W1  = (const float*)d_in[2];
  const float* as1 = (const float*)d_in[3];
  const float* ad1 = (const float*)d_in[4];
  const float* b1  = (const float*)d_in[5];
  const float* g1  = (const float*)d_in[6];
  const float* be1 = (const float*)d_in[7];
  const float* m1  = (const float*)d_in[8];
  const float* v1  = (const float*)d_in[9];
  const float* W2  = (const float*)d_in[10];
  const float* as2 = (const float*)d_in[11];
  const float* ad2 = (const float*)d_in[12];
  const float* b2  = (const float*)d_in[13];
  const float* g2  = (const float*)d_in[14];
  const float* be2 = (const float*)d_in[15];
  const float* m2  = (const float*)d_in[16];
  const float* v2  = (const float*)d_in[17];

  const int N    = in_sizes[0] / 128;  // 50000 (multiple of 16)
  const int E    = in_sizes[1] / 2;    // 800000
  const int Etot = E + N;

  // workspace carve-out (~213 MB)
  char* ws = (char*)d_ws;
  size_t off = 0;
  auto carve = [&](size_t bytes) {
    void* p = ws + off;
    off = (off + bytes + 255) & ~(size_t)255;
    return p;
  };
  float*    h1   = (float*)carve((size_t)N * 256 * 4);
  float*    acc1 = (float*)carve((size_t)N * 256 * 4);
  float*    h2   = (float*)carve((size_t)N * 512 * 4);
  float*    es   = (float*)carve((size_t)N * 2 * 4);
  float*    ed   = (float*)carve((size_t)N * 2 * 4);
  unsigned* mkey = (unsigned*)carve((size_t)N * 2 * 4);
  float*    ssum = (float*)carve((size_t)N * 2 * 4);
  float*    ew   = (float*)carve((size_t)Etot * 2 * 4);
  (void)ws_size; (void)n_in; (void)out_size;

  float* out = (float*)d_out;

  const int TB = 256;
  const int eb = (Etot + TB - 1) / TB;
  const int sb = (2 * N + TB - 1) / TB;
  const int ab = (N + 7) / 8;  // attn_logits: 8 waves/block

  // =========================== Layer 1 (HC = 256) ==========================
  {
    dim3 g(N / 16, 256 / 64);
    gemm_wmma_f32<128><<<g, 128, 0, stream>>>(x, W1, h1, 256);

    attn_logits<<<ab, 256, 0, stream>>>(h1, as1, ad1, es, ed, N, 128);
    init_stats<<<sb, TB, 0, stream>>>(mkey, ssum, 2 * N);

    long long tot = (long long)N * 256;
    init_bias<256><<<(unsigned)((tot + TB - 1) / TB), TB, 0, stream>>>(acc1, b1, tot);

    edge_max<<<eb, TB, 0, stream>>>(ei, E, N, es, ed, ew, mkey);
    edge_exp<<<eb, TB, 0, stream>>>(ei, E, N, mkey, ew, ssum);

    long long work = (long long)Etot * 256;
    edge_scatter<256><<<(unsigned)((work + TB - 1) / TB), TB, 0, stream>>>(
        ei, E, N, ew, ssum, h1, acc1);

    bn_relu<256><<<(unsigned)((tot + TB - 1) / TB), TB, 0, stream>>>(
        acc1, g1, be1, m1, v1, tot);
  }

  // =========================== Layer 2 (HC = 512) ==========================
  {
    dim3 g(N / 16, 512 / 64);
    gemm_wmma_f32<256><<<g, 128, 0, stream>>>(acc1, W2, h2, 512);

    attn_logits<<<ab, 256, 0, stream>>>(h2, as2, ad2, es, ed, N, 256);
    init_stats<<<sb, TB, 0, stream>>>(mkey, ssum, 2 * N);

    long long tot = (long long)N * 512;
    init_bias<512><<<(unsigned)((tot + TB - 1) / TB), TB, 0, stream>>>(out, b2, tot);

    edge_max<<<eb, TB, 0, stream>>>(ei, E, N, es, ed, ew, mkey);
    edge_exp<<<eb, TB, 0, stream>>>(ei, E, N, mkey, ew, ssum);

    long long work = (long long)Etot * 512;
    edge_scatter<512><<<(unsigned)((work + TB - 1) / TB), TB, 0, stream>>>(
        ei, E, N, ew, ssum, h2, out);

    bn_relu<512><<<(unsigned)((tot + TB - 1) / TB), TB, 0, stream>>>(
        out, g2, be2, m2, v2, tot);
  }
}